// FCOSPostProcessor_71081708749426
// MI455X (gfx1250) — compile-verified
//
#include <hip/hip_runtime.h>
#include <math.h>

// ---------------------------------------------------------------------------
// FCOS post-processor for MI455X (gfx1250, wave32).
//   Stage 1: histogram-based top-500 per (img, level) + box decode   (BW bound)
//   Stage 2: per-image bitonic sort of 2560 candidates in LDS
//   Stage 3: pairwise-IoU suppression bit-matrix, 16x16 tiles per wave:
//            - boxes staged global->LDS with async-tensor path (ASYNCcnt)
//            - area_i + area_j outer-sum via V_WMMA_F32_16X16X4_F32
//            - per-tile bits gathered with wave32 ballot
//   Stage 4: sequential greedy scan of the bit matrix (matches reference
//            order semantics exactly), rank + scatter top-100.
// ---------------------------------------------------------------------------

#define N_IMG   16
#define N_CLS   80
#define N_LEV   5
#define KPER    500
#define K       2500
#define KPAD    2560          // 160 tiles of 16
#define MASKW   (KPAD / 32)   // 80 words per row
#define NBIN    4096
#define SORTN   4096
#define POST_N  100
#define IMG_WF  1216.0f
#define IMG_HF  800.0f
#define PRE_THRESH 0.05f
#define TAU0   -2.9444389791664403f   // logit(0.05)
#define NMS_T   0.6f
#define CLS_OFF 4096.0f

typedef __attribute__((ext_vector_type(2))) float v2f;
typedef __attribute__((ext_vector_type(8))) float v8f;
typedef __attribute__((ext_vector_type(4))) int   v4i;

#if __has_builtin(__builtin_amdgcn_wmma_f32_16x16x4_f32)
#define HAVE_WMMA4 1
#endif
#if __has_builtin(__builtin_amdgcn_global_load_async_to_lds_b128)
#define USE_ASYNC_LDS 1
#endif

// Builtin signature (from clang diagnostic): arg0 is v4i in addrspace(1)
// (printed as "__device__"), arg1 the LDS-side v4i pointer, then imm offset
// and imm cpol.
typedef __attribute__((address_space(1))) v4i* gptr_b128;
typedef __attribute__((address_space(3))) v4i* lptr_b128;

static __device__ __forceinline__ gptr_b128 to_gbl128(const void* p) {
  return (gptr_b128)(unsigned long long)p;
}
static __device__ __forceinline__ lptr_b128 to_lds128(void* p) {
  // generic LDS pointer -> 32-bit LDS offset (low dword of generic address)
  return (lptr_b128)(unsigned int)(unsigned long long)p;
}

struct Lv {
  const float* cls[N_LEV];
  const float* reg[N_LEV];
  const float* loc[N_LEV];
};

static __device__ __forceinline__ int lev_hw(int l) {
  switch (l) {
    case 0: return 15200;  // 100x152
    case 1: return 3800;   // 50x76
    case 2: return 950;    // 25x38
    case 3: return 228;    // 12x19
    default: return 54;    // 6x9
  }
}

// ---------------------------------------------------------------------------
// Stage 0: init candidate arrays (unfilled slots behave like masked -1 entries)
// ---------------------------------------------------------------------------
__global__ void k_init(float4* cbox, float* cscore, int* clabel) {
  int g = blockIdx.x * blockDim.x + threadIdx.x;
  if (g < N_IMG * KPAD) {
    cbox[g]   = make_float4(0.f, 0.f, 0.f, 0.f);
    cscore[g] = -1.0f;
    clabel[g] = 0;
  }
}

// ---------------------------------------------------------------------------
// Stage 1a: per (img, level) logit histogram (only logits > logit(0.05))
// ---------------------------------------------------------------------------
__global__ void __launch_bounds__(256) k_hist(Lv lv, unsigned* hist) {
  int lev = blockIdx.y, img = blockIdx.z;
  int hw  = lev_hw(lev);
  int n   = hw * N_CLS;
  const float* p = lv.cls[lev] + (size_t)img * n;

  __shared__ unsigned h[NBIN];
  for (int i = threadIdx.x; i < NBIN; i += 256) h[i] = 0;
  __syncthreads();

  int chunk = (n + gridDim.x - 1) / gridDim.x;
  int s = blockIdx.x * chunk;
  int e = min(n, s + chunk);
  for (int i = s + (int)threadIdx.x; i < e; i += 256) {
    float l = p[i];
    if (l > TAU0) {
      int b = (int)((l + 3.0f) * (NBIN / 16.0f));   // bins over [-3, 13)
      b = b < 0 ? 0 : (b > NBIN - 1 ? NBIN - 1 : b);
      atomicAdd(&h[b], 1u);
    }
  }
  __syncthreads();
  unsigned* gh = hist + (size_t)(img * N_LEV + lev) * NBIN;
  for (int i = threadIdx.x; i < NBIN; i += 256) {
    unsigned v = h[i];
    if (v) atomicAdd(&gh[i], v);
  }
}

// ---------------------------------------------------------------------------
// Stage 1b: find threshold bin so that (strictly-above count) < 500 <= +bin
// sel4 layout per (img,level): {tbin, countAbove, ctrHi, ctrEq}
// ---------------------------------------------------------------------------
__global__ void k_select(const unsigned* hist, int* sel4) {
  if (threadIdx.x != 0) return;
  int il = blockIdx.x;
  const unsigned* h = hist + (size_t)il * NBIN;
  unsigned cum = 0; int tbin = -1; int above = 0;
  for (int b = NBIN - 1; b >= 0; --b) {
    unsigned c = h[b];
    if (cum + c >= (unsigned)KPER) { tbin = b; above = (int)cum; break; }
    cum += c;
  }
  sel4[il * 4 + 0] = tbin;   // -1 => take everything (fewer than 500 pass)
  sel4[il * 4 + 1] = above;
}

// ---------------------------------------------------------------------------
// Stage 1c: collect top-500, decode + clip boxes
// ---------------------------------------------------------------------------
__global__ void __launch_bounds__(256) k_collect(Lv lv, int* sel4,
                                                 float4* cbox, float* cscore,
                                                 int* clabel) {
  int lev = blockIdx.y, img = blockIdx.z;
  int hw  = lev_hw(lev);
  int n   = hw * N_CLS;
  int il  = img * N_LEV + lev;
  const float* p  = lv.cls[lev] + (size_t)img * n;
  const float* rg = lv.reg[lev] + (size_t)img * 4 * hw;
  const float* lc = lv.loc[lev];
  int tbin  = sel4[il * 4 + 0];
  int above = sel4[il * 4 + 1];
  int* cHi  = &sel4[il * 4 + 2];
  int* cEq  = &sel4[il * 4 + 3];

  int chunk = (n + gridDim.x - 1) / gridDim.x;
  int s = blockIdx.x * chunk;
  int e = min(n, s + chunk);
  for (int i = s + (int)threadIdx.x; i < e; i += 256) {
    float l = p[i];
    if (!(l > TAU0)) continue;
    int b = (int)((l + 3.0f) * (NBIN / 16.0f));
    b = b < 0 ? 0 : (b > NBIN - 1 ? NBIN - 1 : b);
    int pos;
    if (b > tbin)       pos = atomicAdd(cHi, 1);            // guaranteed < above
    else if (b == tbin) pos = above + atomicAdd(cEq, 1);
    else continue;
    if (pos >= KPER) continue;

    int c = i / hw, q = i - c * hw;        // memory layout (C, H*W)
    float sc = 1.0f / (1.0f + expf(-l));
    float px = lc[q * 2], py = lc[q * 2 + 1];
    float r0 = rg[q], r1 = rg[hw + q], r2 = rg[2 * hw + q], r3 = rg[3 * hw + q];
    float x1 = fminf(fmaxf(px - r0, 0.f), IMG_WF - 1.f);
    float y1 = fminf(fmaxf(py - r1, 0.f), IMG_HF - 1.f);
    float x2 = fminf(fmaxf(px + r2, 0.f), IMG_WF - 1.f);
    float y2 = fminf(fmaxf(py + r3, 0.f), IMG_HF - 1.f);
    bool valid = (sc > PRE_THRESH) && (x2 - x1 + 1.f >= 0.f) && (y2 - y1 + 1.f >= 0.f);
    int slot = img * KPAD + lev * KPER + pos;
    cbox[slot]   = make_float4(x1, y1, x2, y2);
    cscore[slot] = valid ? sc : -1.0f;     // masked score, like reference
    clabel[slot] = c + 1;
  }
}

// ---------------------------------------------------------------------------
// Stage 2: per-image bitonic sort (desc score, asc index) of 2560 candidates
// ---------------------------------------------------------------------------
__global__ void __launch_bounds__(512) k_sort(const float* cscore,
                                              const float4* cbox,
                                              const int* clabel,
                                              float4* sbx, float4* sobox,
                                              float* ssc, int* slb) {
  __shared__ float sk[SORTN];
  __shared__ int   sv[SORTN];
  int img = blockIdx.x, t = threadIdx.x;
  for (int g = t; g < SORTN; g += 512) {
    sk[g] = (g < KPAD) ? cscore[(size_t)img * KPAD + g] : -3.0f;
    sv[g] = g;
  }
  __syncthreads();
  for (int k = 2; k <= SORTN; k <<= 1) {
    for (int j = k >> 1; j > 0; j >>= 1) {
      for (int i = t; i < SORTN; i += 512) {
        int x = i ^ j;
        if (x > i) {
          float a = sk[i], b = sk[x];
          int va = sv[i], vb = sv[x];
          bool aGTb = (a > b) || (a == b && va < vb);
          bool desc = ((i & k) == 0);
          if (desc ? !aGTb : aGTb) {
            sk[i] = b; sk[x] = a; sv[i] = vb; sv[x] = va;
          }
        }
      }
      __syncthreads();
    }
  }
  for (int p2 = t; p2 < KPAD; p2 += 512) {
    int src = sv[p2];
    size_t si = (size_t)img * KPAD + src;
    size_t di = (size_t)img * KPAD + p2;
    float4 b = cbox[si];
    int lab  = clabel[si];
    float ofv = (float)lab * CLS_OFF;      // class-aware NMS offset
    sbx[di]   = b;
    sobox[di] = make_float4(b.x + ofv, b.y + ofv, b.z + ofv, b.w + ofv);
    ssc[di]   = sk[p2];
    slb[di]   = lab;
  }
}

// ---------------------------------------------------------------------------
// Stage 3: 16x16 IoU tile -> 16-bit row masks via wmma outer-sum + ballot
// ---------------------------------------------------------------------------
static __device__ __forceinline__ void tile16(const float4* sb, int rowbase,
                                              int colbase, int lane,
                                              const float4* rb, const float* ra,
                                              float areaRowLane, unsigned* m) {
  int  jl = lane & 15;
  bool lo = lane < 16;
  int  rsel = rowbase + ((lane >> 4) << 3);
  float4 cb = sb[colbase + jl];
  float  ca = (cb.z - cb.x + 1.f) * (cb.w - cb.y + 1.f);
#ifdef HAVE_WMMA4
  // D[i][j] = area_row[i]*1 + 1*area_col[j]  (16x16x4 f32 wmma, K=2 used)
  v2f A; A.x = lo ? areaRowLane : 0.f; A.y = lo ? 1.f : 0.f;   // A[m][0],A[m][1]
  v2f B; B.x = lo ? 1.f : 0.f;         B.y = lo ? ca  : 0.f;   // B[0][n],B[1][n]
  v8f C = {0.f, 0.f, 0.f, 0.f, 0.f, 0.f, 0.f, 0.f};
  v8f D = __builtin_amdgcn_wmma_f32_16x16x4_f32(false, A, false, B, (short)0, C,
                                                false, false);
#endif
#pragma unroll
  for (int r = 0; r < 8; ++r) {
#ifdef HAVE_WMMA4
    float ps = D[r];                 // area_i + area_j for this lane's (i,j)
#else
    float ps = ra[r] + ca;
#endif
    float ltx = fmaxf(rb[r].x, cb.x);
    float lty = fmaxf(rb[r].y, cb.y);
    float rbx = fminf(rb[r].z, cb.z);
    float rby = fminf(rb[r].w, cb.w);
    float iw = fmaxf(rbx - ltx + 1.f, 0.f);
    float ih = fmaxf(rby - lty + 1.f, 0.f);
    float inter = iw * ih;
    float uni = ps - inter;          // union > 0 always (areas >= 1)
    bool pred = (inter > NMS_T * uni) && ((colbase + jl) > (rsel + r));
    m[r] = __builtin_amdgcn_ballot_w32(pred);
  }
}

__global__ void __launch_bounds__(256) k_mask(const float4* __restrict__ sobox,
                                              unsigned* __restrict__ maskb) {
  __shared__ float4 sb[KPAD];        // 40 KB of 320 KB WGP LDS
  int img = blockIdx.y, rowtile = blockIdx.x;
  int tid = threadIdx.x;
  const float4* src = sobox + (size_t)img * KPAD;

  // Stage whole sorted (offset) box array into LDS: async global->LDS path
  for (int g = tid; g < KPAD; g += 256) {
#ifdef USE_ASYNC_LDS
    __builtin_amdgcn_global_load_async_to_lds_b128(to_gbl128(src + g),
                                                   to_lds128(&sb[g]), 0, 0);
#else
    sb[g] = src[g];
#endif
  }
#ifdef USE_ASYNC_LDS
  asm volatile("s_wait_asynccnt 0" ::: "memory");
#endif
  __syncthreads();

  int lane = tid & 31, wave = tid >> 5;
  int rowbase = rowtile * 16;
  int jl = lane & 15;
  int rsel = rowbase + ((lane >> 4) << 3);

  float4 rb[8]; float ra[8];
#pragma unroll
  for (int r = 0; r < 8; ++r) {
    rb[r] = sb[rsel + r];
    ra[r] = (rb[r].z - rb[r].x + 1.f) * (rb[r].w - rb[r].y + 1.f);
  }
  float4 arow = sb[rowbase + jl];    // lane<16 supplies A operand rows
  float areaRowLane = (arow.z - arow.x + 1.f) * (arow.w - arow.y + 1.f);

  for (int ct = wave; ct < MASKW; ct += 8) {   // 80/8 = 10 iters, uniform
    unsigned mA[8], mB[8];
    tile16(sb, rowbase, ct * 32,      lane, rb, ra, areaRowLane, mA);
    tile16(sb, rowbase, ct * 32 + 16, lane, rb, ra, areaRowLane, mB);
    if (lane == 0) {
#pragma unroll
      for (int rho = 0; rho < 16; ++rho) {
        unsigned ba = (rho < 8) ? (mA[rho] & 0xffffu) : (mA[rho - 8] >> 16);
        unsigned bb = (rho < 8) ? (mB[rho] & 0xffffu) : (mB[rho - 8] >> 16);
        maskb[(size_t)(img * KPAD + rowbase + rho) * MASKW + ct] = ba | (bb << 16);
      }
    }
  }
}

// ---------------------------------------------------------------------------
// Stage 4: sequential greedy suppression over the bit matrix, rank, scatter
// ---------------------------------------------------------------------------
__global__ void __launch_bounds__(128) k_scan(const float4* sbx, const float* ssc,
                                              const int* slb,
                                              const unsigned* maskb,
                                              float* outb, float* outs,
                                              float* outl) {
  __shared__ unsigned keepw[MASKW];
  __shared__ unsigned wscan[MASKW + 1];
  int img = blockIdx.x, t = threadIdx.x;

  if (t < MASKW) {
    unsigned w = 0;
    for (int b = 0; b < 32; ++b) {
      float s = ssc[(size_t)img * KPAD + t * 32 + b];
      if (s > PRE_THRESH) w |= (1u << b);       // valid_s
    }
    keepw[t] = w;
  }
  __syncthreads();

  for (int i = 0; i < K; ++i) {
    bool kb = (keepw[i >> 5] >> (i & 31)) & 1u;
    __syncthreads();
    if (kb && t < MASKW)
      keepw[t] &= ~maskb[(size_t)(img * KPAD + i) * MASKW + t];
    __syncthreads();
  }

  if (t == 0) {
    unsigned c = 0;
    for (int w2 = 0; w2 < MASKW; ++w2) { wscan[w2] = c; c += __popc(keepw[w2]); }
    wscan[MASKW] = c;
  }
  __syncthreads();

  for (int g = t; g < K; g += 128) {
    unsigned w = keepw[g >> 5];
    if ((w >> (g & 31)) & 1u) {
      unsigned rank = wscan[g >> 5] + __popc(w & ((1u << (g & 31)) - 1u));
      if (rank < POST_N) {
        float4 b = sbx[(size_t)img * KPAD + g];
        size_t o = (size_t)img * POST_N + rank;
        outb[o * 4 + 0] = b.x; outb[o * 4 + 1] = b.y;
        outb[o * 4 + 2] = b.z; outb[o * 4 + 3] = b.w;
        outs[o] = ssc[(size_t)img * KPAD + g];
        outl[o] = (float)slb[(size_t)img * KPAD + g];
      }
    }
  }
}

// ---------------------------------------------------------------------------
// Host launch (graph-capture safe: only async memset + kernels on `stream`)
// Workspace use: ~17.3 MB
// ---------------------------------------------------------------------------
extern "C" void kernel_launch(void* const* d_in, const int* in_sizes, int n_in,
                              void* d_out, int out_size, void* d_ws,
                              size_t ws_size, hipStream_t stream) {
  (void)in_sizes; (void)n_in; (void)ws_size;

  Lv lv;
  for (int l = 0; l < N_LEV; ++l) {   // setup_inputs dict order: cls,reg,loc per level
    lv.cls[l] = (const float*)d_in[3 * l + 0];
    lv.reg[l] = (const float*)d_in[3 * l + 1];
    lv.loc[l] = (const float*)d_in[3 * l + 2];
  }

  char* W = (char*)d_ws;
  size_t off = 0;
  auto take = [&](size_t bytes) -> void* {
    void* p = W + off;
    off = (off + bytes + 255) & ~(size_t)255;
    return p;
  };
  unsigned* hist  = (unsigned*)take((size_t)N_IMG * N_LEV * NBIN * 4);
  int*      sel4  = (int*)     take((size_t)N_IMG * N_LEV * 16);
  float4*   cbox  = (float4*)  take((size_t)N_IMG * KPAD * 16);
  float*    cscore= (float*)   take((size_t)N_IMG * KPAD * 4);
  int*      clabel= (int*)     take((size_t)N_IMG * KPAD * 4);
  float4*   sbx   = (float4*)  take((size_t)N_IMG * KPAD * 16);
  float4*   sobox = (float4*)  take((size_t)N_IMG * KPAD * 16);
  float*    ssc   = (float*)   take((size_t)N_IMG * KPAD * 4);
  int*      slb   = (int*)     take((size_t)N_IMG * KPAD * 4);
  unsigned* maskb = (unsigned*)take((size_t)N_IMG * KPAD * MASKW * 4);

  (void)hipMemsetAsync(hist, 0, (size_t)N_IMG * N_LEV * NBIN * 4, stream);
  (void)hipMemsetAsync(sel4, 0, (size_t)N_IMG * N_LEV * 16, stream);
  (void)hipMemsetAsync(d_out, 0, (size_t)out_size * sizeof(float), stream);

  k_init<<<dim3((N_IMG * KPAD + 255) / 256), 256, 0, stream>>>(cbox, cscore, clabel);
  k_hist<<<dim3(32, N_LEV, N_IMG), 256, 0, stream>>>(lv, hist);
  k_select<<<dim3(N_IMG * N_LEV), 32, 0, stream>>>(hist, sel4);
  k_collect<<<dim3(32, N_LEV, N_IMG), 256, 0, stream>>>(lv, sel4, cbox, cscore, clabel);
  k_sort<<<dim3(N_IMG), 512, 0, stream>>>(cscore, cbox, clabel, sbx, sobox, ssc, slb);
  k_mask<<<dim3(KPAD / 16, N_IMG), 256, 0, stream>>>(sobox, maskb);

  float* outb = (float*)d_out;
  float* outs = outb + (size_t)N_IMG * POST_N * 4;
  float* outl = outs + (size_t)N_IMG * POST_N;
  k_scan<<<dim3(N_IMG), 128, 0, stream>>>(sbx, ssc, slb, maskb, outb, outs, outl);
}